// CLAPP_SNN_16295105921260
// MI455X (gfx1250) — compile-verified
//
#include <hip/hip_runtime.h>

// CLAPP SNN on MI455X (gfx1250): bandwidth-bound fused matvec + rank-1 update
// passes using V_WMMA_F32_16X16X4_F32 (wave32).
//
// Per hidden layer per step:
//   fc pass   : apply deferred rank-1 update(s) from previous step, write W back,
//               matvec with updated W -> cur; epilogue does LIF + u1 = LR*bf*fb*sg(cur)
//   retro pass: matvec with OLD retro_w -> retrod, apply rank-1 (spk x fb), write back;
//               epilogue u2 = bf*retrod*sg(cur_prev)
//   pred pass : apply rank-1 (fb x spk), write back, matvec with NEW pred_w -> fb_next
//   out pass  : fc_out matvec + LIF, writes spk_rec / mem_rec slices of d_out
//
// Inner loop uses B128 loads: matvec result is invariant under K-permutation,
// and the same permutation is applied to A (weights) and B (vector), so one
// contiguous float4 per lane feeds two WMMAs with no cross-lane exchange.

typedef __attribute__((ext_vector_type(2))) float v2f;
typedef __attribute__((ext_vector_type(4))) float v4f;
typedef __attribute__((ext_vector_type(8))) float v8f;

#define HID   4096
#define KD    4096
#define OUTD  512
#define NT    64
#define LRv   2e-6f

__device__ __forceinline__ float surrogate_f(float x) {
    const float PI = 3.14159265358979323846f;
    float px = PI * x;
    return 1.0f / (PI * (1.0f + px * px));
}

// One wave: 16 rows x [kb,ke) of K. Per iteration a lane loads a contiguous
// float4 of its row at k + 4*lanehi (non-temporal: W is streamed) and the
// matching float4 of the LDS-staged vectors; two WMMAs consume it.
template<int NUP, int USENEW>
__device__ __forceinline__ v8f tile_loop(const float* __restrict__ Wrow,
                                         float* __restrict__ WrowOut,
                                         const float* lx, const float* lv1,
                                         const float* lv2,
                                         float a1, float a2,
                                         int kb, int ke, int lanehi)
{
    v8f acc = {0.f, 0.f, 0.f, 0.f, 0.f, 0.f, 0.f, 0.f};
    const int lo = (lanehi << 2);
    const v4f* __restrict__ wp = (const v4f*)(Wrow + kb + lo);
    v4f* __restrict__ wo       = WrowOut ? (v4f*)(WrowOut + kb + lo) : nullptr;
    const v4f* xp  = (const v4f*)(lx + kb + lo);
    const v4f* c1p = (const v4f*)(lv1 + kb + lo);
    const v4f* c2p = (const v4f*)(lv2 + kb + lo);
    const int niter = (ke - kb) >> 3;        // K advances 8 per iteration
    #pragma unroll 4
    for (int it = 0; it < niter; ++it) {
        v4f w  = __builtin_nontemporal_load(wp + 2 * it);
        v4f xb = xp[2 * it];
        v4f wm = w;                          // matvec value (old W)
        if (NUP >= 1) {
            v4f c1 = c1p[2 * it];
            w.x = fmaf(a1, c1.x, w.x);
            w.y = fmaf(a1, c1.y, w.y);
            w.z = fmaf(a1, c1.z, w.z);
            w.w = fmaf(a1, c1.w, w.w);
            if (NUP >= 2) {
                v4f c2 = c2p[2 * it];
                w.x = fmaf(a2, c2.x, w.x);
                w.y = fmaf(a2, c2.y, w.y);
                w.z = fmaf(a2, c2.z, w.z);
                w.w = fmaf(a2, c2.w, w.w);
            }
            __builtin_nontemporal_store(w, wo + 2 * it);
            if (USENEW) wm = w;              // matvec uses updated W
        }
        // Two WMMAs per float4; identical K-permutation applied to A and B.
        v2f A0 = {wm.x, wm.y}, B0 = {xb.x, xb.y};
        acc = __builtin_amdgcn_wmma_f32_16x16x4_f32(false, A0, false, B0,
                                                    (short)0, acc, false, false);
        v2f A1 = {wm.z, wm.w}, B1 = {xb.z, xb.w};
        acc = __builtin_amdgcn_wmma_f32_16x16x4_f32(false, A1, false, B1,
                                                    (short)0, acc, false, false);
    }
    return acc;
}

// Generic fused pass. Block = 128 threads (4 wave32), 16 rows per block,
// K=4096 split 1024 per wave, LDS reduction, per-row epilogue.
// epi_mode: 1 = hidden fc+LIF, 2 = retro(u2), 3 = pred(fb out), 4 = output LIF
__global__ __launch_bounds__(128) void clapp_matpass(
    const float* __restrict__ W, float* __restrict__ Wout, int n_rows,
    const float* __restrict__ xB,
    const float* __restrict__ v1, const float* __restrict__ v2,
    const float* __restrict__ rs1, const float* __restrict__ rs2,
    int scale_lr, int nup, int use_new,
    const float* __restrict__ bias, int epi_mode,
    float* ep_cur, float* ep_mem, const float* ep_fb,
    float* ep_spk, float* ep_u1,
    const float* ep_curprev, float* ep_u2,
    float* ep_fbout,
    float* ep_spkrec, float* ep_memrec,
    const int* __restrict__ bf_ptr, int t)
{
    __shared__ alignas(16) float lx[KD];
    __shared__ alignas(16) float lv1[KD];
    __shared__ alignas(16) float lv2[KD];
    __shared__ float red[4][16];

    const int tid    = threadIdx.x;
    const int wave   = tid >> 5;
    const int lane   = tid & 31;
    const int lanelo = lane & 15;
    const int lanehi = lane >> 4;
    const int row    = blockIdx.x * 16 + lanelo;

    // Stage the shared vectors once per workgroup (16KB each).
    {
        const v4f* sx = (const v4f*)xB;
        v4f* dx = (v4f*)lx;
        for (int i = tid; i < KD / 4; i += 128) dx[i] = sx[i];
        if (nup >= 1) {
            const v4f* s1 = (const v4f*)v1;
            v4f* d1 = (v4f*)lv1;
            for (int i = tid; i < KD / 4; i += 128) d1[i] = s1[i];
        }
        if (nup >= 2) {
            const v4f* s2 = (const v4f*)v2;
            v4f* d2 = (v4f*)lv2;
            for (int i = tid; i < KD / 4; i += 128) d2[i] = s2[i];
        }
    }
    __syncthreads();

    const float bfv = (t == 0) ? (float)bf_ptr[0] : 1.0f;
    const float sA  = scale_lr ? (LRv * bfv) : 1.0f;
    const float a1  = (nup >= 1) ? sA * rs1[row] : 0.0f;
    const float a2  = (nup >= 2) ? sA * rs2[row] : 0.0f;

    const size_t rowoff = (size_t)row * (size_t)KD;
    const float* Wrow   = W + rowoff;
    float* WrowOut      = Wout ? (Wout + rowoff) : nullptr;
    const int kb = wave * (KD / 4);
    const int ke = kb + (KD / 4);

    v8f acc;
    if (nup == 0)
        acc = tile_loop<0, 0>(Wrow, WrowOut, lx, lv1, lv2, a1, a2, kb, ke, lanehi);
    else if (nup == 1 && !use_new)
        acc = tile_loop<1, 0>(Wrow, WrowOut, lx, lv1, lv2, a1, a2, kb, ke, lanehi);
    else if (nup == 1)
        acc = tile_loop<1, 1>(Wrow, WrowOut, lx, lv1, lv2, a1, a2, kb, ke, lanehi);
    else
        acc = tile_loop<2, 1>(Wrow, WrowOut, lx, lv1, lv2, a1, a2, kb, ke, lanehi);

    // D layout: VGPR r, lane L -> M = r + 8*(L>>4), N = L&15 (all N identical).
    if (lanelo == 0) {
        #pragma unroll
        for (int r = 0; r < 8; ++r) red[wave][(lanehi << 3) + r] = acc[r];
    }
    __syncthreads();

    if (tid < 16) {
        const int orow = blockIdx.x * 16 + tid;
        float cur = red[0][tid] + red[1][tid] + red[2][tid] + red[3][tid];
        if (bias) cur += bias[orow];
        switch (epi_mode) {
        case 1: { // hidden fc: record cur, LIF, compute next-step u1
            ep_cur[orow] = cur;
            float mem   = ep_mem[orow];
            float reset = (mem > 1.0f) ? 1.0f : 0.0f;
            mem = 0.95f * mem + cur - reset;
            float spk = (mem > 1.0f) ? 1.0f : 0.0f;
            ep_mem[orow] = mem;
            ep_spk[orow] = spk;
            ep_u1[orow]  = LRv * bfv * ep_fb[orow] * surrogate_f(cur);
        } break;
        case 2: { // retro: cur == retrod
            ep_u2[orow] = bfv * cur * surrogate_f(ep_curprev[orow]);
        } break;
        case 3: { // pred: cur == new feedback
            ep_fbout[orow] = cur;
        } break;
        case 4: { // output LIF -> d_out
            float mem   = ep_mem[orow];
            float reset = (mem > 1.0f) ? 1.0f : 0.0f;
            mem = 0.95f * mem + cur - reset;
            float spk = (mem > 1.0f) ? 1.0f : 0.0f;
            ep_mem[orow]    = mem;
            ep_spkrec[orow] = spk;
            ep_memrec[orow] = mem;
        } break;
        }
    }
}

__global__ void clapp_zero(float* p, int n) {
    int i = blockIdx.x * blockDim.x + threadIdx.x;
    if (i < n) p[i] = 0.0f;
}

extern "C" void kernel_launch(void* const* d_in, const int* in_sizes, int n_in,
                              void* d_out, int out_size, void* d_ws, size_t ws_size,
                              hipStream_t stream) {
    (void)in_sizes; (void)n_in; (void)out_size; (void)ws_size;
    const float* inp       = (const float*)d_in[0];   // 64 x 4096
    const int*   bf        = (const int*)d_in[1];
    const float* l1_fc_w   = (const float*)d_in[2];
    const float* l1_fc_b   = (const float*)d_in[3];
    const float* l1_pred_w = (const float*)d_in[4];
    const float* l1_retro_w= (const float*)d_in[5];
    const float* l2_fc_w   = (const float*)d_in[6];
    const float* l2_fc_b   = (const float*)d_in[7];
    const float* l2_pred_w = (const float*)d_in[8];
    const float* l2_retro_w= (const float*)d_in[9];
    const float* fco_w     = (const float*)d_in[10];  // 512 x 4096
    const float* fco_b     = (const float*)d_in[11];
    float* out = (float*)d_out;

    const size_t WSZ = (size_t)HID * (size_t)KD;      // 16.7M floats per matrix
    float* ws  = (float*)d_ws;
    float* W1f = ws + 0 * WSZ;
    float* W1p = ws + 1 * WSZ;
    float* W1r = ws + 2 * WSZ;
    float* W2f = ws + 3 * WSZ;
    float* W2p = ws + 4 * WSZ;
    float* W2r = ws + 5 * WSZ;

    // Inputs must not be mutated: work on private copies (graph-capture-safe).
    hipMemcpyAsync(W1f, l1_fc_w,    WSZ * sizeof(float), hipMemcpyDeviceToDevice, stream);
    hipMemcpyAsync(W1p, l1_pred_w,  WSZ * sizeof(float), hipMemcpyDeviceToDevice, stream);
    hipMemcpyAsync(W1r, l1_retro_w, WSZ * sizeof(float), hipMemcpyDeviceToDevice, stream);
    hipMemcpyAsync(W2f, l2_fc_w,    WSZ * sizeof(float), hipMemcpyDeviceToDevice, stream);
    hipMemcpyAsync(W2p, l2_pred_w,  WSZ * sizeof(float), hipMemcpyDeviceToDevice, stream);
    hipMemcpyAsync(W2r, l2_retro_w, WSZ * sizeof(float), hipMemcpyDeviceToDevice, stream);

    // State (zero-initialized every call: recurrence starts from zeros).
    float* st = ws + 6 * WSZ;
    float* base = st;
    auto alloc = [&](size_t n) { float* p = st; st += n; return p; };
    float* cur1 = alloc(2 * HID);  // double buffer (cur / cur_prev)
    float* fb1  = alloc(2 * HID);  // double buffer (feedback old / new)
    float* mem1 = alloc(HID);
    float* u1_1 = alloc(HID);      // pending LR*bf*fb*sg(cur)
    float* u2_1 = alloc(HID);      // pending bf*retrod*sg(cur_prev)
    float* spk1 = alloc(3 * HID);  // triple buffer: layer2 inputs t, t-1, t-2
    float* cur2 = alloc(2 * HID);
    float* fb2  = alloc(2 * HID);
    float* mem2 = alloc(HID);
    float* u1_2 = alloc(HID);
    float* u2_2 = alloc(HID);
    float* spkh = alloc(HID);      // layer2 spikes (current step only)
    float* memo = alloc(OUTD);
    const int nstate = (int)(st - base);
    clapp_zero<<<(nstate + 255) / 256, 256, 0, stream>>>(base, nstate);

    for (int t = 0; t < NT; ++t) {
        const float* x_t  = inp + (size_t)t * KD;
        const float* x_m1 = (t >= 1) ? inp + (size_t)(t - 1) * KD : nullptr;
        const float* x_m2 = (t >= 2) ? inp + (size_t)(t - 2) * KD : nullptr;
        const int nup = (t == 0) ? 0 : ((t == 1) ? 1 : 2);

        float* c1t = cur1 + (t & 1) * HID;
        float* c1p = cur1 + ((t + 1) & 1) * HID;
        float* f1t = fb1 + (t & 1) * HID;
        float* f1n = fb1 + ((t + 1) & 1) * HID;
        float* s1t  = spk1 + (t % 3) * HID;
        float* s1m1 = spk1 + ((t + 2) % 3) * HID;  // spk1(t-1)
        float* s1m2 = spk1 + ((t + 1) % 3) * HID;  // spk1(t-2)
        float* c2t = cur2 + (t & 1) * HID;
        float* c2p = cur2 + ((t + 1) & 1) * HID;
        float* f2t = fb2 + (t & 1) * HID;
        float* f2n = fb2 + ((t + 1) & 1) * HID;

        // ---- Layer 1 ----
        // fc: deferred dW(t-1) = u1*x(t-1)^T + u2*x(t-2)^T, matvec(new W, x_t), LIF
        clapp_matpass<<<HID / 16, 128, 0, stream>>>(
            W1f, W1f, HID, x_t, x_m1, x_m2, u1_1, u2_1, /*scale_lr=*/0, nup, /*use_new=*/1,
            l1_fc_b, /*epi=*/1, c1t, mem1, f1t, s1t, u1_1,
            nullptr, nullptr, nullptr, nullptr, nullptr, bf, t);
        // retro: retrod = oldW @ spk ; W += LR*bf * spk x fb ; u2 = bf*retrod*sg(cur_prev)
        clapp_matpass<<<HID / 16, 128, 0, stream>>>(
            W1r, W1r, HID, s1t, f1t, nullptr, s1t, nullptr, /*scale_lr=*/1, 1, /*use_new=*/0,
            nullptr, /*epi=*/2, nullptr, nullptr, nullptr, nullptr, nullptr,
            c1p, u2_1, nullptr, nullptr, nullptr, bf, t);
        // pred: W += LR*bf * fb x spk ; fb_next = newW @ spk
        clapp_matpass<<<HID / 16, 128, 0, stream>>>(
            W1p, W1p, HID, s1t, s1t, nullptr, f1t, nullptr, /*scale_lr=*/1, 1, /*use_new=*/1,
            nullptr, /*epi=*/3, nullptr, nullptr, nullptr, nullptr, nullptr,
            nullptr, nullptr, f1n, nullptr, nullptr, bf, t);

        // ---- Layer 2 (input = spk1) ----
        clapp_matpass<<<HID / 16, 128, 0, stream>>>(
            W2f, W2f, HID, s1t, s1m1, s1m2, u1_2, u2_2, /*scale_lr=*/0, nup, /*use_new=*/1,
            l2_fc_b, /*epi=*/1, c2t, mem2, f2t, spkh, u1_2,
            nullptr, nullptr, nullptr, nullptr, nullptr, bf, t);
        clapp_matpass<<<HID / 16, 128, 0, stream>>>(
            W2r, W2r, HID, spkh, f2t, nullptr, spkh, nullptr, /*scale_lr=*/1, 1, /*use_new=*/0,
            nullptr, /*epi=*/2, nullptr, nullptr, nullptr, nullptr, nullptr,
            c2p, u2_2, nullptr, nullptr, nullptr, bf, t);
        clapp_matpass<<<HID / 16, 128, 0, stream>>>(
            W2p, W2p, HID, spkh, spkh, nullptr, f2t, nullptr, /*scale_lr=*/1, 1, /*use_new=*/1,
            nullptr, /*epi=*/3, nullptr, nullptr, nullptr, nullptr, nullptr,
            nullptr, nullptr, f2n, nullptr, nullptr, bf, t);

        // ---- Output layer: cur = fc_out_w @ spkh + b ; LIF -> d_out ----
        clapp_matpass<<<OUTD / 16, 128, 0, stream>>>(
            fco_w, nullptr, OUTD, spkh, nullptr, nullptr, nullptr, nullptr, 0, 0, 0,
            fco_b, /*epi=*/4, nullptr, memo, nullptr, nullptr, nullptr,
            nullptr, nullptr, nullptr,
            out + (size_t)t * OUTD, out + (size_t)NT * OUTD + (size_t)t * OUTD, bf, t);
    }
}